// BiasEncoderDense_12335146074377
// MI455X (gfx1250) — compile-verified
//
#include <hip/hip_runtime.h>

typedef __attribute__((ext_vector_type(2))) float v2f;
typedef __attribute__((ext_vector_type(8))) float v8f;

#define NHEADS   32
#define NB       8
#define NN       256
#define NP1      257
#define DDIM     3
#define KTOT     96          // DDIM * NHEADS, GEMM reduction dim
#define EPAD     98          // padded e-row stride (conflict-free b64 loads)
#define TJ       16          // points (j's) per wave tile
#define WAVES    8
#define THREADS  256
#define NSTEP    (KTOT / 4)  // 24 chained 16x16x4 steps

// One wave computes a 32(heads) x 16(points) output tile:
//   C = W'(32x96) @ e^T(96x16), chained V_WMMA_F32_16X16X4_F32, K=96.
__global__ __launch_bounds__(THREADS)
void bias_inner_kernel(const float* __restrict__ attn_bias,
                       const float* __restrict__ spatial_W,
                       const float* __restrict__ edge_W,
                       const float* __restrict__ edge_dis_W,
                       const int*   __restrict__ spatial_pos,
                       const int*   __restrict__ edge_input,
                       float*       __restrict__ out)
{
    // Fragment-ready W': Wsw[((s*2+m)*32 + lane)*2 + v] = W'[m*16+(l&15)][4s+v+koff]
    __shared__ float Wsw[NSTEP * 2 * 32 * 2];    // 12 KB
    __shared__ float eLds[WAVES][TJ * EPAD];     // ~49 KB: per-wave e, [point][K] padded

    const int tid  = threadIdx.x;
    const int lane = tid & 31;
    const int wave = tid >> 5;

    // tile id -> (b, i, j-tile); 16 j-tiles per row, 256 rows per batch
    const int tile = blockIdx.x * WAVES + wave;
    const int jt   = tile & 15;
    const int i    = (tile >> 4) & 255;
    const int b    = tile >> 12;
    const int j0   = jt * TJ;

    // stage Wd (edge_dis_W first 96*32 floats) in WMMA-fragment order:
    // each lane's A fragment becomes one contiguous ds_load_b64
    for (int idx = tid; idx < NSTEP * 2 * 32 * 2; idx += THREADS) {
        const int v = idx & 1;
        const int l = (idx >> 1) & 31;
        const int m = (idx >> 6) & 1;
        const int s = idx >> 7;
        const int K = 4 * s + v + ((l >> 4) << 1);
        const int h = m * 16 + (l & 15);
        Wsw[idx] = edge_dis_W[K * NHEADS + h];
    }

    // build e^T staging: c encodes (point p, hop d); lane = head h.
    // All lanes share the two edge indices (broadcast); edge_W gather is a
    // coalesced 128B row. padding_idx==0 rows contribute zero.
    {
        const int base_bi = (b * NN + i) * NN;
        #pragma unroll 4
        for (int c = 0; c < TJ * DDIM; ++c) {
            const int p  = c / DDIM;
            const int d  = c - DDIM * p;
            const int eb = (base_bi + (j0 + p)) * (DDIM * 2) + d * 2;
            const int i0 = edge_input[eb];
            const int i1 = edge_input[eb + 1];
            float v = 0.0f;
            if (i0 != 0) v += edge_W[i0 * NHEADS + lane];
            if (i1 != 0) v += edge_W[i1 * NHEADS + lane];
            eLds[wave][p * EPAD + d * NHEADS + lane] = 0.5f * v;
        }
    }
    __syncthreads();

    // --- WMMA accumulation: K = 96 via 24 chained 16x16x4 steps, 2 M-tiles ---
    // 16x4 f32 A/B lane layout: lanes 0-15 hold K=4s+{0,1}, lanes 16-31 K=4s+{2,3}
    const int hi   = lane >> 4;          // lane-half
    const int hl   = lane & 15;
    const int koff = hi * 2;

    v8f c0 = {};                         // heads 0..15  x points
    v8f c1 = {};                         // heads 16..31 x points
    const float* eme = &eLds[wave][hl * EPAD];   // this lane's point column of e^T
    const float* wme = &Wsw[lane * 2];

    #pragma unroll
    for (int s = 0; s < NSTEP; ++s) {
        const v2f bfrag = *(const v2f*)&eme[4 * s + koff];   // e^T row-pair, col hl
        const v2f a0    = *(const v2f*)&wme[s * 128];        // m=0 fragment
        const v2f a1    = *(const v2f*)&wme[s * 128 + 64];   // m=1 fragment
        c0 = __builtin_amdgcn_wmma_f32_16x16x4_f32(false, a0, false, bfrag,
                                                   (short)0, c0, false, false);
        c1 = __builtin_amdgcn_wmma_f32_16x16x4_f32(false, a1, false, bfrag,
                                                   (short)0, c1, false, false);
    }

    // --- epilogue: per-lane point p = hl is fixed across all 16 heads ---
    const int p    = hl;
    const int spos = spatial_pos[(b * NN + i) * NN + j0 + p];
    int sp = (spos == 0) ? 1 : spos;
    if (sp > 1) sp -= 1;
    if (sp > DDIM) sp = DDIM;
    const float rsp = 1.0f / (float)sp;
    const float ab2 = 2.0f * attn_bias[(b * NP1 + (i + 1)) * NP1 + (j0 + p + 1)];

    // D layout: VGPR v, lanes 0-15 -> M=v, lanes 16-31 -> M=v+8; N = hl
    const int obase = ((b * NHEADS) * NP1 + (i + 1)) * NP1 + (j0 + p + 1);
    #pragma unroll
    for (int v = 0; v < 8; ++v) {
        {
            const int head = v + hi * 8;
            const float spat = (spos != 0) ? spatial_W[spos * NHEADS + head] : 0.0f;
            out[obase + head * (NP1 * NP1)] = c0[v] * rsp + spat + ab2;
        }
        {
            const int head = 16 + v + hi * 8;
            const float spat = (spos != 0) ? spatial_W[spos * NHEADS + head] : 0.0f;
            out[obase + head * (NP1 * NP1)] = c1[v] * rsp + spat + ab2;
        }
    }
}

// Row 0 (all j) and column 0 (i>=1): 2*attn_bias + token_W[h]
__global__ __launch_bounds__(THREADS)
void bias_border_kernel(const float* __restrict__ attn_bias,
                        const float* __restrict__ token_W,
                        float*       __restrict__ out)
{
    const int per_bh = NP1 + NN;         // 257 row elems + 256 col elems
    int t = blockIdx.x * blockDim.x + threadIdx.x;
    if (t >= NB * NHEADS * per_bh) return;
    const int bh = t / per_bh;
    const int r  = t - bh * per_bh;
    const int b  = bh / NHEADS;
    const int h  = bh - b * NHEADS;
    const float tk = token_W[h];
    if (r < NP1) {
        const int j = r;
        out[((b * NHEADS + h) * NP1 + 0) * NP1 + j] =
            2.0f * attn_bias[(b * NP1 + 0) * NP1 + j] + tk;
    } else {
        const int ii = r - NP1 + 1;      // 1..256
        out[((b * NHEADS + h) * NP1 + ii) * NP1 + 0] =
            2.0f * attn_bias[(b * NP1 + ii) * NP1 + 0] + tk;
    }
}

extern "C" void kernel_launch(void* const* d_in, const int* in_sizes, int n_in,
                              void* d_out, int out_size, void* d_ws, size_t ws_size,
                              hipStream_t stream) {
    const float* attn_bias   = (const float*)d_in[0];
    const float* spatial_W   = (const float*)d_in[1];
    const float* edge_W      = (const float*)d_in[2];
    const float* edge_dis_W  = (const float*)d_in[3];
    const float* token_W     = (const float*)d_in[4];
    const int*   spatial_pos = (const int*)d_in[5];
    const int*   edge_input  = (const int*)d_in[6];
    float* out = (float*)d_out;

    // 8 batches * 256 rows * 16 j-tiles = 32768 wave-tiles, 8 waves/block
    bias_inner_kernel<<<4096, THREADS, 0, stream>>>(
        attn_bias, spatial_W, edge_W, edge_dis_W, spatial_pos, edge_input, out);

    // 8*32*513 border elements
    const int nb = (NB * NHEADS * (NP1 + NN) + THREADS - 1) / THREADS;
    bias_border_kernel<<<nb, THREADS, 0, stream>>>(attn_bias, token_W, out);
}